// EntityNLM_26130581029606
// MI455X (gfx1250) — compile-verified
//
#include <hip/hip_runtime.h>
#include <cstdint>
#include <math.h>

#define H   128
#define T   512
#define V   50257
#define E   64
#define G4  512        // 4*H
#define VE  (V + E)

typedef __attribute__((ext_vector_type(2))) float v2f;
typedef __attribute__((ext_vector_type(8))) float v8f;

// ---------------------------------------------------------------------------
// Shared f32 WMMA 16x16 tile (K=128) using V_WMMA_F32_16X16X4_F32 (exact f32).
// A-frag (16x4): lane&15 = M row, lane>>4 selects K pair {0,1}/{2,3}.
// B-frag (4x16): lane&15 = N col, same K-pair select (B passed as row n of W,
// since C = A @ W^T  =>  B[k][n] = W[n][k]).
// C/D:  VGPR r -> M = r (lanes 0-15) / M = 8+r (lanes 16-31), N = lane&15.
// ---------------------------------------------------------------------------
__device__ __forceinline__ v8f tile_gemm_k128(const float* __restrict__ arow,
                                              const float* __restrict__ brow,
                                              bool bvalid) {
  const int kh = ((threadIdx.x >> 4) & 1) << 1;
  v8f c = {};
#pragma unroll
  for (int k0 = 0; k0 < H; k0 += 4) {
    v2f a, b;
    a.x = arow[k0 + kh];
    a.y = arow[k0 + kh + 1];
    float b0 = brow[k0 + kh];
    float b1 = brow[k0 + kh + 1];
    b.x = bvalid ? b0 : 0.0f;
    b.y = bvalid ? b1 : 0.0f;
    c = __builtin_amdgcn_wmma_f32_16x16x4_f32(false, a, false, b,
                                              (short)0, c, false, false);
  }
  return c;
}

// ---------------------------------------------------------------------------
// Kernel 1: Gx[t][j] = W_ih @ embed[tokens[t]] + b_ih[j] + b_hh[j]
// GEMM [512 x 128] (gathered rows) @ [128 x 512] -> [512 x 512]
// 32 M-tiles x 32 N-tiles = 1024 tiles, 8 waves/block -> 128 blocks.
// ---------------------------------------------------------------------------
__global__ __launch_bounds__(256) void k_gx(const int* __restrict__ tokens,
                                            const float* __restrict__ embed,
                                            const float* __restrict__ W_ih,
                                            const float* __restrict__ b_ih,
                                            const float* __restrict__ b_hh,
                                            float* __restrict__ Gx) {
  const int lane = threadIdx.x & 31;
  const int wid  = threadIdx.x >> 5;
  const int tile = blockIdx.x * 8 + wid;         // 0..1023
  const int m0 = (tile & 31) * 16;
  const int n0 = (tile >> 5) * 16;
  const int m  = m0 + (lane & 15);
  const int n  = n0 + (lane & 15);
  const float* arow = embed + (size_t)tokens[m] * H;
  const float* brow = W_ih + (size_t)n * H;
  v8f c = tile_gemm_k128(arow, brow, true);
  const int   mb   = m0 + ((lane >> 4) & 1) * 8;
  const float bias = b_ih[n] + b_hh[n];
#pragma unroll
  for (int r = 0; r < 8; ++r)
    Gx[(size_t)(mb + r) * G4 + n] = c[r] + bias;
}

// ---------------------------------------------------------------------------
// Kernel 2: sequential LSTM. 1 block, 512 threads (16 waves on one WGP).
// Thread j owns gate j: W_hh row j lives in 32 float4 VGPRs for all 512 steps.
// Gx rows are double-buffered into LDS with async global->LDS loads
// (ASYNCcnt path), prefetching step t+1 while computing step t.
// ---------------------------------------------------------------------------
__device__ __forceinline__ void async_gx_row(const float* __restrict__ src,
                                             float* dst, int tid) {
  if (tid < 128) {
    uint32_t lds_off = (uint32_t)(uintptr_t)(dst + tid * 4);
    unsigned long long ga = (unsigned long long)(uintptr_t)(src + tid * 4);
    asm volatile("global_load_async_to_lds_b128 %0, %1, off"
                 :: "v"(lds_off), "v"(ga) : "memory");
  }
}
__device__ __forceinline__ void async_wait_all() {
  asm volatile("s_wait_asynccnt 0" ::: "memory");
}

__global__ __launch_bounds__(512) void k_lstm(const float* __restrict__ Gx,
                                              const float* __restrict__ W_hh,
                                              float* __restrict__ Hst) {
  __shared__ __align__(16) float h_s[H];
  __shared__ float g_s[G4];
  __shared__ __align__(16) float gx_s[2][G4];
  const int tid = threadIdx.x;

  // Register-cache this thread's W_hh row (128 f32 = 32 float4 VGPR quads).
  float4 wrow[32];
  const float4* wsrc = (const float4*)(W_hh + (size_t)tid * H);
#pragma unroll
  for (int i = 0; i < 32; ++i) wrow[i] = wsrc[i];

  if (tid < H) h_s[tid] = 0.0f;
  float cstate = 0.0f;

  async_gx_row(Gx, gx_s[0], tid);   // stage Gx[0]
  async_wait_all();
  __syncthreads();

  for (int t = 0; t < T; ++t) {
    const int cur = t & 1;
    if (t + 1 < T) async_gx_row(Gx + (size_t)(t + 1) * G4, gx_s[cur ^ 1], tid);

    // gate[tid] = Gx[t][tid] + W_hh[tid,:] . h   (h broadcast from LDS)
    float acc = gx_s[cur][tid];
    const float4* hv4 = (const float4*)h_s;
#pragma unroll
    for (int i = 0; i < 32; ++i) {
      float4 w = wrow[i];
      float4 h4 = hv4[i];
      acc += w.x * h4.x + w.y * h4.y + w.z * h4.z + w.w * h4.w;
    }
    g_s[tid] = acc;
    __syncthreads();

    if (tid < H) {
      float ig = 1.0f / (1.0f + __expf(-g_s[tid]));
      float fg = 1.0f / (1.0f + __expf(-g_s[tid + H]));
      float gg = tanhf(g_s[tid + 2 * H]);
      float og = 1.0f / (1.0f + __expf(-g_s[tid + 3 * H]));
      cstate = fg * cstate + ig * gg;
      float hh = og * tanhf(cstate);
      h_s[tid] = hh;
      Hst[(size_t)t * H + tid] = hh;
    }
    async_wait_all();   // next step's Gx buffer landed in LDS
    __syncthreads();
  }
}

// ---------------------------------------------------------------------------
// Kernel 3: Ae = H @ W_e^T, Ad = H @ W_delta^T   ([512,128]@[128,128] x2)
// 2 mats x 32 M-tiles x 8 N-tiles = 512 tiles -> 64 blocks x 8 waves.
// ---------------------------------------------------------------------------
__global__ __launch_bounds__(256) void k_eproj(const float* __restrict__ Hst,
                                               const float* __restrict__ W_e,
                                               const float* __restrict__ W_delta,
                                               float* __restrict__ Ae,
                                               float* __restrict__ Ad) {
  const int lane = threadIdx.x & 31;
  const int wid  = threadIdx.x >> 5;
  const int tile = blockIdx.x * 8 + wid;         // 0..511
  const int mat  = tile >> 8;
  const int tt   = tile & 255;
  const int m0 = (tt & 31) * 16;
  const int n0 = (tt >> 5) * 16;
  const int m  = m0 + (lane & 15);
  const int n  = n0 + (lane & 15);
  const float* W = mat ? W_delta : W_e;
  float*       O = mat ? Ad : Ae;
  v8f c = tile_gemm_k128(Hst + (size_t)m * H, W + (size_t)n * H, true);
  const int mb = m0 + ((lane >> 4) & 1) * 8;
#pragma unroll
  for (int r = 0; r < 8; ++r)
    O[(size_t)(mb + r) * H + n] = c[r];
}

// ---------------------------------------------------------------------------
// Kernel 4: sequential entity module. 1 block, 256 threads.
// ents (64x128) live in LDS; per step: pred_e, bilinear gate d, blend,
// renormalize, scatter. Wave32 __shfl_down reductions.
// ---------------------------------------------------------------------------
__global__ __launch_bounds__(256) void k_entity(const int* __restrict__ eids,
                                                const float* __restrict__ sids,
                                                const float* __restrict__ ents0,
                                                const float* __restrict__ dist0,
                                                const float* __restrict__ Hst,
                                                const float* __restrict__ Ae,
                                                const float* __restrict__ Ad,
                                                const float* __restrict__ b_e,
                                                const float* __restrict__ w_dw,
                                                const float* __restrict__ w_db,
                                                const float* __restrict__ b_delta,
                                                float* __restrict__ out) {
  __shared__ float ents_s[E * H];
  __shared__ float dist_s[E];
  __shared__ float red[256];
  __shared__ float sc[2];
  const int tid = threadIdx.x;
  const int e = tid >> 2, q = tid & 3;

  // load + row-normalize initial entities (create_entity)
  float ss = 0.0f;
  for (int k = q * 32; k < q * 32 + 32; ++k) {
    float v = ents0[e * H + k];
    ents_s[e * H + k] = v;
    ss += v * v;
  }
  red[tid] = ss;
  if (tid < E) dist_s[tid] = dist0[tid];
  __syncthreads();
  float inv0 = rsqrtf(red[e * 4] + red[e * 4 + 1] + red[e * 4 + 2] + red[e * 4 + 3]);
  for (int k = q * 32; k < q * 32 + 32; ++k) ents_s[e * H + k] *= inv0;
  __syncthreads();

  const float be = b_e[0], wdw = w_dw[0], wdb = w_db[0], bd = b_delta[0];

  for (int t = 0; t < T; ++t) {
    const int   eid = eids[t];
    const float sid = sids[t];
    const float* ae = Ae + (size_t)t * H;
    const float* ad = Ad + (size_t)t * H;
    const float* hh = Hst + (size_t)t * H;

    // pred_e[e] = ents[e,:] . (W_e h) + b_e + (dist[e]-sid)*w_dw + w_db
    float p = 0.0f;
    for (int k = q * 32; k < q * 32 + 32; ++k) p += ents_s[e * H + k] * ae[k];
    red[tid] = p;
    __syncthreads();
    if (q == 0) {
      float s = red[tid] + red[tid + 1] + red[tid + 2] + red[tid + 3];
      out[(size_t)t * VE + V + e] = s + be + (dist_s[e] - sid) * wdw + wdb;
    }
    __syncthreads();

    // d = sigmoid(ents[eid,:] . (W_delta h) + b_delta)
    float pd = (tid < H) ? ents_s[eid * H + tid] * ad[tid] : 0.0f;
    for (int off = 16; off > 0; off >>= 1) pd += __shfl_down(pd, off, 32);
    if ((tid & 31) == 0) red[tid >> 5] = pd;
    __syncthreads();
    if (tid == 0) {
      float s = red[0] + red[1] + red[2] + red[3];
      sc[0] = 1.0f / (1.0f + __expf(-(s + bd)));
    }
    __syncthreads();

    // blend + renormalize + scatter
    const float d = sc[0];
    float ne = 0.0f, pn = 0.0f;
    if (tid < H) {
      ne = d * ents_s[eid * H + tid] + (1.0f - d) * hh[tid];
      pn = ne * ne;
    }
    for (int off = 16; off > 0; off >>= 1) pn += __shfl_down(pn, off, 32);
    if ((tid & 31) == 0) red[(tid >> 5) + 8] = pn;
    __syncthreads();
    if (tid == 0) sc[1] = rsqrtf(red[8] + red[9] + red[10] + red[11]);
    __syncthreads();
    if (tid < H) ents_s[eid * H + tid] = ne * sc[1];
    if (tid == 0) dist_s[eid] = sid;
    __syncthreads();
  }
}

// ---------------------------------------------------------------------------
// Kernel 5: pred_x = H @ W_x^T + b_x   ([512,128] @ [128,50257])
// 32 M-tiles x 3142 N-tiles = 100544 tiles -> 12568 blocks x 8 waves.
// Ragged last N-tile masked in B-loads and stores. Write-BW bound (103 MB).
// ---------------------------------------------------------------------------
__global__ __launch_bounds__(256) void k_predx(const float* __restrict__ Hst,
                                               const float* __restrict__ W_x,
                                               const float* __restrict__ b_x,
                                               float* __restrict__ out) {
  const int lane = threadIdx.x & 31;
  const int wid  = threadIdx.x >> 5;
  const int tile = blockIdx.x * 8 + wid;         // 0..100543
  const int m0 = (tile & 31) * 16;
  const int nT = tile >> 5;                       // 0..3141
  const int n0 = nT * 16;
  const int m  = m0 + (lane & 15);
  const int n  = n0 + (lane & 15);
  const bool nv = (n < V);
  const float* arow = Hst + (size_t)m * H;
  const float* brow = W_x + (size_t)(nv ? n : (V - 1)) * H;
  __builtin_prefetch(brow, 0, 3);                 // global_prefetch_b8
  v8f c = tile_gemm_k128(arow, brow, nv);
  const int mb = m0 + ((lane >> 4) & 1) * 8;
  if (nv) {
    const float bias = b_x[n];
#pragma unroll
    for (int r = 0; r < 8; ++r)
      out[(size_t)(mb + r) * VE + n] = c[r] + bias;
  }
}

// ---------------------------------------------------------------------------
extern "C" void kernel_launch(void* const* d_in, const int* in_sizes, int n_in,
                              void* d_out, int out_size, void* d_ws, size_t ws_size,
                              hipStream_t stream) {
  const int*   tokens  = (const int*)  d_in[0];
  const int*   eids    = (const int*)  d_in[1];
  const float* sids    = (const float*)d_in[2];
  const float* ents0   = (const float*)d_in[3];
  const float* dist0   = (const float*)d_in[4];
  const float* embed   = (const float*)d_in[5];
  const float* W_ih    = (const float*)d_in[6];
  const float* W_hh    = (const float*)d_in[7];
  const float* b_ih    = (const float*)d_in[8];
  const float* b_hh    = (const float*)d_in[9];
  const float* W_x     = (const float*)d_in[10];
  const float* b_x     = (const float*)d_in[11];
  const float* W_e     = (const float*)d_in[12];
  const float* b_e     = (const float*)d_in[13];
  const float* w_dw    = (const float*)d_in[14];
  const float* w_db    = (const float*)d_in[15];
  const float* W_delta = (const float*)d_in[16];
  const float* b_delta = (const float*)d_in[17];

  float* out = (float*)d_out;
  float* Gx  = (float*)d_ws;                       // [T][4H]   1 MB
  float* Hst = Gx  + (size_t)T * G4;               // [T][H]  256 KB
  float* Ae  = Hst + (size_t)T * H;                // [T][H]  256 KB
  float* Ad  = Ae  + (size_t)T * H;                // [T][H]  256 KB

  k_gx    <<<128,   256, 0, stream>>>(tokens, embed, W_ih, b_ih, b_hh, Gx);
  k_lstm  <<<1,     512, 0, stream>>>(Gx, W_hh, Hst);
  k_eproj <<<64,    256, 0, stream>>>(Hst, W_e, W_delta, Ae, Ad);
  k_entity<<<1,     256, 0, stream>>>(eids, sids, ents0, dist0, Hst, Ae, Ad,
                                      b_e, w_dw, w_db, b_delta, out);
  k_predx <<<12568, 256, 0, stream>>>(Hst, W_x, b_x, out);
}